// EAMForceCUDAExt_11854109737007
// MI455X (gfx1250) — compile-verified
//
#include <hip/hip_runtime.h>
#include <stdint.h>

// ---- compile-time table shape (matches reference; N/E taken from in_sizes) ----
#define NTYPES 2
#define NR     8192
#define NRHO   4096

static constexpr float BOX_L   = 10.0f;
static constexpr float INV_BOX = 1.0f / 10.0f;
static constexpr float CUTOFF  = 6.0f;
static constexpr float EPSF    = 1e-7f;
static constexpr float INV_DR  = 8191.0f / 6.0f;              // (NR-1)/CUTOFF
static constexpr float RMAXC   = 6.0f * (1.0f - 1e-7f);       // CUTOFF*(1-EPS)

#define EDGE_BLOCKS 1024
#define TPB 256

// ---------------- CDNA5 helpers ----------------

// Relaxed agent-scope f32 atomic add -> global_atomic_add_f32 (L2 atomic units)
__device__ __forceinline__ void atomAddF32(float* p, float v) {
    __hip_atomic_fetch_add(p, v, __ATOMIC_RELAXED, __HIP_MEMORY_SCOPE_AGENT);
}

// Async global -> LDS copy of 16 bytes (ASYNCcnt-tracked, no VGPR data path).
// VDST is a VGPR holding the LDS byte address (low 32 bits of flat shared addr).
__device__ __forceinline__ void asyncCopyB128ToLds(uint32_t lds_byte_addr, const float* gptr) {
    asm volatile("global_load_async_to_lds_b128 %0, %1, off"
                 :: "v"(lds_byte_addr), "v"((unsigned long long)(uintptr_t)gptr)
                 : "memory");
}
__device__ __forceinline__ void waitAsyncCnt0() {
    asm volatile("s_wait_asynccnt 0" ::: "memory");
}

// Stage NTYPES*NR floats (64 KB) from global into LDS via the async copy path.
__device__ __forceinline__ void stageTableToLds(float* sdst, const float* gsrc) {
    uint32_t base = (uint32_t)(uintptr_t)sdst;            // LDS byte offset
    for (int k = (int)threadIdx.x * 4; k < NTYPES * NR; k += (int)blockDim.x * 4) {
        asyncCopyB128ToLds(base + (uint32_t)(k * 4), gsrc + k);
    }
    waitAsyncCnt0();       // this wave's transfers landed in LDS
    __syncthreads();       // all waves' transfers visible
}

// wave32 reduction (warpSize == 32 on gfx1250)
__device__ __forceinline__ float waveReduceSum(float v) {
    v += __shfl_xor(v, 16, 32);
    v += __shfl_xor(v,  8, 32);
    v += __shfl_xor(v,  4, 32);
    v += __shfl_xor(v,  2, 32);
    v += __shfl_xor(v,  1, 32);
    return v;
}

// minimum-image displacement + radial table index/frac from raw deltas
__device__ __forceinline__ bool geomFromDelta(float& dx, float& dy, float& dz, float& r,
                                              int& idx, int& nxt, float& frac) {
    dx -= rintf(dx * INV_BOX) * BOX_L;   // round-half-even == jnp.round
    dy -= rintf(dy * INV_BOX) * BOX_L;
    dz -= rintf(dz * INV_BOX) * BOX_L;
    float r2 = fmaxf(dx * dx + dy * dy + dz * dz, 1e-24f);
    r = sqrtf(r2);
    float rc   = fminf(fmaxf(r, 0.0f), RMAXC);
    float idxf = rc * INV_DR;
    idx  = (int)idxf;
    frac = idxf - (float)idx;
    nxt  = min(idx + 1, NR - 1);
    return r < CUTOFF;
}

__device__ __forceinline__ bool edgeGeom(const float* __restrict__ coords, int i, int j,
                                         float& dx, float& dy, float& dz, float& r,
                                         int& idx, int& nxt, float& frac) {
    dx = coords[3 * i + 0] - coords[3 * j + 0];
    dy = coords[3 * i + 1] - coords[3 * j + 1];
    dz = coords[3 * i + 2] - coords[3 * j + 2];
    return geomFromDelta(dx, dy, dz, r, idx, nxt, frac);
}

// ---------------- kernels ----------------

__global__ __launch_bounds__(TPB) void k_zero(float* __restrict__ ws2N,
                                              float* __restrict__ out,
                                              int n_ws, int n_out) {
    int t = blockIdx.x * blockDim.x + threadIdx.x;
    if (t < n_ws)  ws2N[t] = 0.0f;
    if (t < n_out) out[t]  = 0.0f;
}

// Prep: pack {x,y,z,type} per atom (one b128 gather per atom later) and
// interleave pair/pair_deriv into float2 entries (one b64 per lookup later).
__global__ __launch_bounds__(TPB) void k_pack(const float* __restrict__ coords,
                                              const int*   __restrict__ types,
                                              const float* __restrict__ pair,
                                              const float* __restrict__ pderiv,
                                              float4*      __restrict__ packed,
                                              float2*      __restrict__ pairComb,
                                              int N) {
    int stride = gridDim.x * blockDim.x;
    for (int a = blockIdx.x * blockDim.x + threadIdx.x; a < N; a += stride) {
        packed[a] = make_float4(coords[3 * a + 0], coords[3 * a + 1], coords[3 * a + 2],
                                __int_as_float(types[a]));
    }
    for (int k = blockIdx.x * blockDim.x + threadIdx.x; k < NTYPES * NTYPES * NR; k += stride) {
        pairComb[k] = make_float2(pair[k], pderiv[k]);
    }
}

// ---- Pass 1 (packed): rho_i += f_{t(j)}(r), rho_j += f_{t(i)}(r) ----
__global__ __launch_bounds__(TPB) void k_density_p(const float4* __restrict__ packed,
                                                   const int*    __restrict__ ei,
                                                   const float*  __restrict__ dens,
                                                   float*        __restrict__ rho,
                                                   int E) {
    __shared__ float sdens[NTYPES * NR];          // 64 KB of 320 KB/WGP
    stageTableToLds(sdens, dens);

    int stride = gridDim.x * blockDim.x;
    for (int e = blockIdx.x * blockDim.x + threadIdx.x; e < E; e += stride) {
        if (e + stride < E) {                      // global_prefetch_b8 next tile
            __builtin_prefetch(ei + e + stride);
            __builtin_prefetch(ei + E + e + stride);
        }
        int i = ei[e], j = ei[E + e];
        float4 pi = packed[i];                     // one global_load_b128 per atom
        float4 pj = packed[j];
        float dx = pi.x - pj.x, dy = pi.y - pj.y, dz = pi.z - pj.z;
        float r, frac; int idx, nxt;
        if (!geomFromDelta(dx, dy, dz, r, idx, nxt, frac)) continue;

        int ti = __float_as_int(pi.w), tj = __float_as_int(pj.w);
        float a0 = sdens[tj * NR + idx], a1 = sdens[tj * NR + nxt];
        float b0 = sdens[ti * NR + idx], b1 = sdens[ti * NR + nxt];
        atomAddF32(&rho[i], a0 + frac * (a1 - a0));   // f_col
        atomAddF32(&rho[j], b0 + frac * (b1 - b0));   // f_row
    }
}

// ---- Pass 2: per-atom embedding F(rho), dF/drho; energy += sum(F) ----
__global__ __launch_bounds__(TPB) void k_embed(const int*   __restrict__ types,
                                               const float* __restrict__ rho,
                                               const float* __restrict__ emb,
                                               const float* __restrict__ embd,
                                               const float* __restrict__ rmin_t,
                                               const float* __restrict__ invd_t,
                                               float*       __restrict__ dF,
                                               float*       __restrict__ energy,
                                               int N) {
    int a = blockIdx.x * blockDim.x + threadIdx.x;
    float Fv = 0.0f;
    if (a < N) {
        int   t    = types[a];
        float rmin = rmin_t[t];
        float invd = invd_t[t];
        float hi   = rmin + ((float)(NRHO - 1) / invd) * (1.0f - EPSF);
        float rc   = fminf(fmaxf(rho[a], rmin), hi);
        float idxf = (rc - rmin) * invd;
        int   idx  = (int)idxf;
        float frac = idxf - (float)idx;
        int   nxt  = min(idx + 1, NRHO - 1);
        const float* et  = emb  + t * NRHO;
        const float* edt = embd + t * NRHO;
        float e0 = et[idx];
        Fv = e0 + frac * (et[nxt] - e0);
        float d0 = edt[idx];
        dF[a] = d0 + frac * (edt[nxt] - d0);
    }
    float s = waveReduceSum(Fv);
    if ((threadIdx.x & 31u) == 0u) atomAddF32(energy, s);
}

// ---- Pass 3 (packed): pair + embedding forces, energy, virial ----
__global__ __launch_bounds__(TPB) void k_force_p(const float4* __restrict__ packed,
                                                 const int*    __restrict__ ei,
                                                 const float*  __restrict__ densd,
                                                 const float2* __restrict__ pairComb,
                                                 const float*  __restrict__ dF,
                                                 float*        __restrict__ forces,
                                                 float*        __restrict__ energy,
                                                 float*        __restrict__ virial,
                                                 int E) {
    __shared__ float sdd[NTYPES * NR];            // density-deriv table in LDS
    stageTableToLds(sdd, densd);

    float esum = 0.0f, vsum = 0.0f;
    int stride = gridDim.x * blockDim.x;
    for (int e = blockIdx.x * blockDim.x + threadIdx.x; e < E; e += stride) {
        if (e + stride < E) {
            __builtin_prefetch(ei + e + stride);
            __builtin_prefetch(ei + E + e + stride);
        }
        int i = ei[e], j = ei[E + e];
        float4 pi = packed[i];
        float4 pj = packed[j];
        float dx = pi.x - pj.x, dy = pi.y - pj.y, dz = pi.z - pj.z;
        float r, frac; int idx, nxt;
        if (!geomFromDelta(dx, dy, dz, r, idx, nxt, frac)) continue;

        int ti = __float_as_int(pi.w), tj = __float_as_int(pj.w);
        const float2* pc = pairComb + (ti * NTYPES + tj) * NR;
        float2 t0 = pc[idx];                       // b64: {phi, dphi} at idx
        float2 t1 = pc[nxt];                       // b64: {phi, dphi} at nxt
        float phi  = t0.x + frac * (t1.x - t0.x);
        float dphi = t0.y + frac * (t1.y - t0.y);
        float di0 = sdd[ti * NR + idx];
        float df_i = di0 + frac * (sdd[ti * NR + nxt] - di0);
        float dj0 = sdd[tj * NR + idx];
        float df_j = dj0 + frac * (sdd[tj * NR + nxt] - dj0);

        float coeff = dphi + dF[i] * df_j + dF[j] * df_i;     // dE/dr (mask==1)
        float c = -coeff / r;
        float fx = c * dx, fy = c * dy, fz = c * dz;

        atomAddF32(&forces[3 * i + 0],  fx);
        atomAddF32(&forces[3 * i + 1],  fy);
        atomAddF32(&forces[3 * i + 2],  fz);
        atomAddF32(&forces[3 * j + 0], -fx);
        atomAddF32(&forces[3 * j + 1], -fy);
        atomAddF32(&forces[3 * j + 2], -fz);

        esum += phi;
        vsum += coeff * r;
    }
    esum = waveReduceSum(esum);
    vsum = waveReduceSum(vsum);
    if ((threadIdx.x & 31u) == 0u) {
        atomAddF32(energy, esum);
        atomAddF32(virial, -0.5f * vsum);
    }
}

// ---- Fallback (unpacked) edge kernels for small workspaces ----
__global__ __launch_bounds__(TPB) void k_density(const float* __restrict__ coords,
                                                 const int*   __restrict__ ei,
                                                 const int*   __restrict__ types,
                                                 const float* __restrict__ dens,
                                                 float*       __restrict__ rho,
                                                 int E) {
    __shared__ float sdens[NTYPES * NR];
    stageTableToLds(sdens, dens);
    int stride = gridDim.x * blockDim.x;
    for (int e = blockIdx.x * blockDim.x + threadIdx.x; e < E; e += stride) {
        int i = ei[e], j = ei[E + e];
        float dx, dy, dz, r, frac; int idx, nxt;
        if (!edgeGeom(coords, i, j, dx, dy, dz, r, idx, nxt, frac)) continue;
        int ti = types[i], tj = types[j];
        float a0 = sdens[tj * NR + idx], a1 = sdens[tj * NR + nxt];
        float b0 = sdens[ti * NR + idx], b1 = sdens[ti * NR + nxt];
        atomAddF32(&rho[i], a0 + frac * (a1 - a0));
        atomAddF32(&rho[j], b0 + frac * (b1 - b0));
    }
}

__global__ __launch_bounds__(TPB) void k_force(const float* __restrict__ coords,
                                               const int*   __restrict__ ei,
                                               const int*   __restrict__ types,
                                               const float* __restrict__ densd,
                                               const float* __restrict__ pair,
                                               const float* __restrict__ pderiv,
                                               const float* __restrict__ dF,
                                               float*       __restrict__ forces,
                                               float*       __restrict__ energy,
                                               float*       __restrict__ virial,
                                               int E) {
    __shared__ float sdd[NTYPES * NR];
    stageTableToLds(sdd, densd);
    float esum = 0.0f, vsum = 0.0f;
    int stride = gridDim.x * blockDim.x;
    for (int e = blockIdx.x * blockDim.x + threadIdx.x; e < E; e += stride) {
        int i = ei[e], j = ei[E + e];
        float dx, dy, dz, r, frac; int idx, nxt;
        if (!edgeGeom(coords, i, j, dx, dy, dz, r, idx, nxt, frac)) continue;
        int ti = types[i], tj = types[j];
        const float* pp = pair   + (ti * NTYPES + tj) * NR;
        const float* pd = pderiv + (ti * NTYPES + tj) * NR;
        float p0 = pp[idx];
        float phi  = p0 + frac * (pp[nxt] - p0);
        float q0 = pd[idx];
        float dphi = q0 + frac * (pd[nxt] - q0);
        float di0 = sdd[ti * NR + idx];
        float df_i = di0 + frac * (sdd[ti * NR + nxt] - di0);
        float dj0 = sdd[tj * NR + idx];
        float df_j = dj0 + frac * (sdd[tj * NR + nxt] - dj0);
        float coeff = dphi + dF[i] * df_j + dF[j] * df_i;
        float c = -coeff / r;
        float fx = c * dx, fy = c * dy, fz = c * dz;
        atomAddF32(&forces[3 * i + 0],  fx);
        atomAddF32(&forces[3 * i + 1],  fy);
        atomAddF32(&forces[3 * i + 2],  fz);
        atomAddF32(&forces[3 * j + 0], -fx);
        atomAddF32(&forces[3 * j + 1], -fy);
        atomAddF32(&forces[3 * j + 2], -fz);
        esum += phi;
        vsum += coeff * r;
    }
    esum = waveReduceSum(esum);
    vsum = waveReduceSum(vsum);
    if ((threadIdx.x & 31u) == 0u) {
        atomAddF32(energy, esum);
        atomAddF32(virial, -0.5f * vsum);
    }
}

// ---------------- launcher ----------------
extern "C" void kernel_launch(void* const* d_in, const int* in_sizes, int n_in,
                              void* d_out, int out_size, void* d_ws, size_t ws_size,
                              hipStream_t stream) {
    const float* coords = (const float*)d_in[0];
    const int*   ei     = (const int*)  d_in[1];
    const int*   types  = (const int*)  d_in[2];
    const float* dens   = (const float*)d_in[3];
    const float* densd  = (const float*)d_in[4];
    const float* pair   = (const float*)d_in[5];
    const float* pderiv = (const float*)d_in[6];
    const float* emb    = (const float*)d_in[7];
    const float* embd   = (const float*)d_in[8];
    const float* rmin_t = (const float*)d_in[9];
    const float* invd_t = (const float*)d_in[10];

    const int N = in_sizes[2];          // atoms (atom_types count)
    const int E = in_sizes[1] / 2;      // edges (edge_index is [2,E])

    float* out    = (float*)d_out;
    float* energy = out;                // [0]
    float* forces = out + 1;            // [1 .. 3N]
    float* virial = out + out_size - 1; // [3N+1]

    // workspace layout (all 16B-aligned for N multiple of 4):
    //   rho[N] | dF[N] | packed float4[N] | pairComb float2[NT*NT*NR]
    float*  rho      = (float*)d_ws;
    float*  dF       = rho + N;
    float4* packed   = (float4*)(dF + N);
    float2* pairComb = (float2*)(packed + N);
    const size_t needed = (size_t)(2 * N) * 4 + (size_t)N * 16
                        + (size_t)(NTYPES * NTYPES * NR) * 8;
    const bool usePacked = (ws_size >= needed) && ((N & 3) == 0);

    // zero rho, dF and all outputs (harness poisons buffers; never rely on state)
    {
        int nz = 2 * N;
        int tot = nz > out_size ? nz : out_size;
        k_zero<<<(tot + TPB - 1) / TPB, TPB, 0, stream>>>(rho, out, nz, out_size);
    }

    if (usePacked) {
        int packWork = N > NTYPES * NTYPES * NR ? N : NTYPES * NTYPES * NR;
        k_pack<<<(packWork + TPB - 1) / TPB, TPB, 0, stream>>>(coords, types, pair, pderiv,
                                                               packed, pairComb, N);
        k_density_p<<<EDGE_BLOCKS, TPB, 0, stream>>>(packed, ei, dens, rho, E);
        k_embed<<<(N + TPB - 1) / TPB, TPB, 0, stream>>>(types, rho, emb, embd,
                                                         rmin_t, invd_t, dF, energy, N);
        k_force_p<<<EDGE_BLOCKS, TPB, 0, stream>>>(packed, ei, densd, pairComb,
                                                   dF, forces, energy, virial, E);
    } else {
        k_density<<<EDGE_BLOCKS, TPB, 0, stream>>>(coords, ei, types, dens, rho, E);
        k_embed<<<(N + TPB - 1) / TPB, TPB, 0, stream>>>(types, rho, emb, embd,
                                                         rmin_t, invd_t, dF, energy, N);
        k_force<<<EDGE_BLOCKS, TPB, 0, stream>>>(coords, ei, types, densd, pair, pderiv,
                                                 dF, forces, energy, virial, E);
    }
}